// BasicReason_65910568125058
// MI455X (gfx1250) — compile-verified
//
#include <hip/hip_runtime.h>
#include <math.h>

typedef __attribute__((ext_vector_type(2))) float v2f;
typedef __attribute__((ext_vector_type(8))) float v8f;
typedef __attribute__((ext_vector_type(4))) unsigned int u32x4;
typedef __attribute__((ext_vector_type(4))) int i32x4;
typedef __attribute__((ext_vector_type(8))) int i32x8;

#define B_ 32
#define T_ 256
#define C_ 128
#define TOPK_ 4
#define NEIGH_ 5

#if defined(__has_include)
#if __has_include(<hip/amd_detail/amd_gfx1250_TDM.h>)
#define TDM_SIX_ARGS 1
#endif
#endif

// ---------------- K1: P = X @ W^T for the three projection weights ----------
// One wave computes a full 16x128 output strip: 8 accumulators, one A-fragment
// load feeds 8 v_wmma_f32_16x16x4_f32 per k-step.
__global__ __launch_bounds__(32)
void proj_wmma(const float* __restrict__ x,
               const float* __restrict__ Wpf, const float* __restrict__ Wns,
               const float* __restrict__ Wv,
               float* __restrict__ pf, float* __restrict__ ns_, float* __restrict__ vv)
{
    const int lane = threadIdx.x;
    const int h  = lane >> 4;
    const int lr = lane & 15;
    const int mt = blockIdx.x, which = blockIdx.z;
    const float* W  = (which == 0) ? Wpf : (which == 1) ? Wns : Wv;
    float*      dst = (which == 0) ? pf  : (which == 1) ? ns_ : vv;

    const float* arow = x + (size_t)(mt * 16 + lr) * C_ + 2 * h;
    const float* brow[8];
#pragma unroll
    for (int nt = 0; nt < 8; ++nt)
        brow[nt] = W + (size_t)(nt * 16 + lr) * C_ + 2 * h;

    v8f acc[8] = {};
#pragma unroll 4
    for (int k = 0; k < C_; k += 4) {
        v2f a = *(const v2f*)(arow + k);
#pragma unroll
        for (int nt = 0; nt < 8; ++nt) {
            v2f b = *(const v2f*)(brow[nt] + k);
            acc[nt] = __builtin_amdgcn_wmma_f32_16x16x4_f32(false, a, false, b,
                                                            (short)0, acc[nt], false, false);
        }
    }
#pragma unroll
    for (int nt = 0; nt < 8; ++nt) {
        float* drow = dst + (size_t)(mt * 16 + 8 * h) * C_ + nt * 16 + lr;
#pragma unroll
        for (int r = 0; r < 8; ++r) drow[(size_t)r * C_] = acc[nt][r];
    }
}

// ---------------- K2: in-place row L2 normalization of pf and ns ------------
__global__ __launch_bounds__(128)
void rownorm(float* __restrict__ pf, float* __restrict__ ns_)
{
    __shared__ float red[128];
    float* base = (blockIdx.y == 0) ? pf : ns_;
    float* row  = base + (size_t)blockIdx.x * C_;
    const int t = threadIdx.x;
    float val = row[t];
    red[t] = val * val;
    __syncthreads();
#pragma unroll
    for (int s = 64; s > 0; s >>= 1) {
        if (t < s) red[t] += red[t + s];
        __syncthreads();
    }
    float inv = 1.0f / fmaxf(sqrtf(red[0]), 1e-12f);
    row[t] = val * inv;
}

// ---------------- K3: Gram matrices G = pf pf^T, N = ns ns^T per batch ------
// One wave computes a 16x64 strip (4 accumulators).
__global__ __launch_bounds__(32)
void gram_wmma(const float* __restrict__ pf, const float* __restrict__ ns_,
               float* __restrict__ G, float* __restrict__ Nm)
{
    const int lane = threadIdx.x;
    const int h  = lane >> 4;
    const int lr = lane & 15;
    const int xt = blockIdx.x, ytg = blockIdx.y;          // ytg: group of 4 y-tiles
    const int b  = blockIdx.z >> 1, which = blockIdx.z & 1;
    const float* src = ((which == 0) ? pf : ns_) + (size_t)b * T_ * C_;
    float*       dst = ((which == 0) ? G  : Nm ) + (size_t)b * T_ * T_;

    const float* arow = src + (size_t)(xt * 16 + lr) * C_ + 2 * h;
    const float* brow[4];
#pragma unroll
    for (int j = 0; j < 4; ++j)
        brow[j] = src + (size_t)((ytg * 4 + j) * 16 + lr) * C_ + 2 * h;

    v8f acc[4] = {};
#pragma unroll 4
    for (int k = 0; k < C_; k += 4) {
        v2f a = *(const v2f*)(arow + k);
#pragma unroll
        for (int j = 0; j < 4; ++j) {
            v2f b2 = *(const v2f*)(brow[j] + k);
            acc[j] = __builtin_amdgcn_wmma_f32_16x16x4_f32(false, a, false, b2,
                                                           (short)0, acc[j], false, false);
        }
    }
#pragma unroll
    for (int j = 0; j < 4; ++j) {
        float* drow = dst + (size_t)(xt * 16 + 8 * h) * T_ + (ytg * 4 + j) * 16 + lr;
#pragma unroll
        for (int r = 0; r < 8; ++r) drow[(size_t)r * T_] = acc[j][r];
    }
}

// ---------------- K4: logits via Gram-gather, mask, softmax -> attn ---------
// 5 contiguous G rows are staged into LDS by the Tensor Data Mover (one 2D
// tile descriptor: 256 x 5, stride T), gathered N rows by per-thread loads.
__global__ __launch_bounds__(256)
void attn_kernel(const float* __restrict__ G, const float* __restrict__ Nm,
                 const int* __restrict__ radj, const int* __restrict__ inxs,
                 const float* __restrict__ v_pf, const float* __restrict__ g_pf,
                 const float* __restrict__ v_ns, const float* __restrict__ g_ns,
                 float* __restrict__ attn)
{
    __shared__ float gs[NEIGH_ * T_];
    __shared__ float nq[TOPK_ * T_];
    __shared__ float red[T_];
    const int x = blockIdx.x, b = blockIdx.y, t = threadIdx.x;

    float wpf[NEIGH_], wns[TOPK_];
    {
        float n2 = 0.f;
#pragma unroll
        for (int i = 0; i < NEIGH_; ++i) n2 += v_pf[i] * v_pf[i];
        float s = g_pf[0] / sqrtf(n2);
#pragma unroll
        for (int i = 0; i < NEIGH_; ++i) wpf[i] = v_pf[i] * s;
        n2 = 0.f;
#pragma unroll
        for (int n = 0; n < TOPK_; ++n) n2 += v_ns[n] * v_ns[n];
        s = g_ns[0] / sqrtf(n2);
#pragma unroll
        for (int n = 0; n < TOPK_; ++n) wns[n] = v_ns[n] * s;
    }

    int sx = x - 2; sx = sx < 0 ? 0 : sx; sx = sx > (T_ - NEIGH_) ? (T_ - NEIGH_) : sx;
    const float* Gb = G  + (size_t)b * T_ * T_;
    const float* Nb = Nm + (size_t)b * T_ * T_;

    // gathered N rows: per-thread loads (indices are data-dependent)
    const int* ix = inxs + ((size_t)b * T_ + x) * TOPK_;
#pragma unroll
    for (int n = 0; n < TOPK_; ++n) nq[n * T_ + t] = Nb[(size_t)ix[n] * T_ + t];

#if __has_builtin(__builtin_amdgcn_tensor_load_to_lds)
    // TDM: load the 5x256 f32 tile (row stride T) starting at Gb[sx*T] into gs.
    if (t < 32) {
        const unsigned long long ga =
            (unsigned long long)(uintptr_t)(Gb + (size_t)sx * T_);
        const unsigned lds_off = (unsigned)(uintptr_t)(&gs[0]);   // addr[31:0] = LDS offset
        u32x4 g0 = { 1u,                                          // count=1 (valid user D#)
                     lds_off,                                     // lds_addr
                     (unsigned)(ga & 0xffffffffu),                // global_addr[31:0]
                     (unsigned)((ga >> 32) & 0x01ffffffu) | (2u << 30) }; // [56:32] | type=2
        i32x8 g1 = { (int)(2u << 16),          // data_size=4B
                     (int)(256u << 16),        // tensor_dim0=256 (bits 63:48)
                     (int)(5u << 16),          // tensor_dim1=5   (bits 111:80 low half)
                     (int)(256u << 16),        // tile_dim0=256   (bits 127:112)
                     5,                        // tile_dim1=5, tile_dim2=0
                     256,                      // tensor_dim0_stride=256 (low 32)
                     0, 0 };
        i32x4 gz = { 0, 0, 0, 0 };
#ifdef TDM_SIX_ARGS
        i32x8 g4 = { 0, 0, 0, 0, 0, 0, 0, 0 };
        __builtin_amdgcn_tensor_load_to_lds(g0, g1, gz, gz, g4, 0);
#else
        __builtin_amdgcn_tensor_load_to_lds(g0, g1, gz, gz, 0);
#endif
#if __has_builtin(__builtin_amdgcn_s_wait_tensorcnt)
        __builtin_amdgcn_s_wait_tensorcnt(0);
#else
        asm volatile("s_wait_tensorcnt 0x0" ::: "memory");
#endif
    }
#else
#pragma unroll
    for (int i = 0; i < NEIGH_; ++i) gs[i * T_ + t] = Gb[(size_t)(sx + i) * T_ + t];
#endif
    __syncthreads();

    const int y = t;
    int sy = y - 2; sy = sy < 0 ? 0 : sy; sy = sy > (T_ - NEIGH_) ? (T_ - NEIGH_) : sy;

    float spf = 0.f;
#pragma unroll
    for (int i = 0; i < NEIGH_; ++i) {
        float m = gs[i * T_ + sy];
#pragma unroll
        for (int j = 1; j < NEIGH_; ++j) m = fmaxf(m, gs[i * T_ + sy + j]);
        spf += wpf[i] * m;
    }
    float sns = 0.f;
#pragma unroll
    for (int n = 0; n < TOPK_; ++n) sns += wns[n] * nq[n * T_ + y];

    const int adj = radj[((size_t)b * T_ + x) * T_ + y];
    float logit = spf + sns + (adj == 0 ? -1e22f : 0.0f);

    red[t] = logit; __syncthreads();
#pragma unroll
    for (int s = 128; s > 0; s >>= 1) {
        if (t < s) red[t] = fmaxf(red[t], red[t + s]);
        __syncthreads();
    }
    float mx = red[0]; __syncthreads();
    float e = __expf(logit - mx);
    red[t] = e; __syncthreads();
#pragma unroll
    for (int s = 128; s > 0; s >>= 1) {
        if (t < s) red[t] += red[t + s];
        __syncthreads();
    }
    attn[((size_t)b * T_ + x) * T_ + y] = e * (1.0f / red[0]);
}

// ---------------- K5: out = attn @ v (per batch) ----------------------------
// One wave computes the full 16x128 output strip (8 accumulators).
__global__ __launch_bounds__(32)
void av_wmma(const float* __restrict__ attn, const float* __restrict__ vv,
             float* __restrict__ out)
{
    const int lane = threadIdx.x;
    const int h  = lane >> 4;
    const int lr = lane & 15;
    const int mt = blockIdx.x, b = blockIdx.z;
    const float* A = attn + (size_t)b * T_ * T_ + (size_t)(mt * 16 + lr) * T_ + 2 * h;
    const float* V = vv   + (size_t)b * T_ * C_;

    v8f acc[8] = {};
#pragma unroll 4
    for (int k = 0; k < T_; k += 4) {
        v2f a = *(const v2f*)(A + k);
        const float* vr0 = V + (size_t)(k + 2 * h) * C_ + lr;
        const float* vr1 = vr0 + C_;
#pragma unroll
        for (int nt = 0; nt < 8; ++nt) {
            v2f bb;
            bb.x = vr0[nt * 16];
            bb.y = vr1[nt * 16];
            acc[nt] = __builtin_amdgcn_wmma_f32_16x16x4_f32(false, a, false, bb,
                                                            (short)0, acc[nt], false, false);
        }
    }
#pragma unroll
    for (int nt = 0; nt < 8; ++nt) {
        float* drow = out + (size_t)b * T_ * C_ + (size_t)(mt * 16 + 8 * h) * C_ + nt * 16 + lr;
#pragma unroll
        for (int r = 0; r < 8; ++r) drow[(size_t)r * C_] = acc[nt][r];
    }
}

extern "C" void kernel_launch(void* const* d_in, const int* in_sizes, int n_in,
                              void* d_out, int out_size, void* d_ws, size_t ws_size,
                              hipStream_t stream)
{
    (void)in_sizes; (void)n_in; (void)out_size; (void)ws_size;
    const float* x    = (const float*)d_in[0];
    const int*   radj = (const int*)  d_in[1];
    const int*   inxs = (const int*)  d_in[2];
    const float* Wpf  = (const float*)d_in[3];
    const float* Wns  = (const float*)d_in[4];
    const float* Wv   = (const float*)d_in[5];
    const float* vpf  = (const float*)d_in[6];
    const float* gpf  = (const float*)d_in[7];
    const float* vns  = (const float*)d_in[8];
    const float* gns  = (const float*)d_in[9];
    float* out = (float*)d_out;

    float* ws = (float*)d_ws;
    const size_t BTC = (size_t)B_ * T_ * C_;   // 1,048,576 floats
    const size_t BTT = (size_t)B_ * T_ * T_;   // 2,097,152 floats
    float* pf   = ws;
    float* ns_  = pf  + BTC;
    float* vv   = ns_ + BTC;
    float* G    = vv  + BTC;
    float* Nm   = G   + BTT;
    float* attn = Nm  + BTT;                   // total 9.4M floats = 36 MB

    proj_wmma  <<<dim3((B_ * T_) / 16, 1, 3),  32,  0, stream>>>(x, Wpf, Wns, Wv, pf, ns_, vv);
    rownorm    <<<dim3(B_ * T_, 2),            128, 0, stream>>>(pf, ns_);
    gram_wmma  <<<dim3(T_ / 16, 4, B_ * 2),    32,  0, stream>>>(pf, ns_, G, Nm);
    attn_kernel<<<dim3(T_, B_),                T_,  0, stream>>>(G, Nm, radj, inxs,
                                                                 vpf, gpf, vns, gns, attn);
    av_wmma    <<<dim3(T_ / 16, 1, B_),        32,  0, stream>>>(attn, vv, out);
}